// BasicRNN_5909874999704
// MI455X (gfx1250) — compile-verified
//
#include <hip/hip_runtime.h>
#include <hip/hip_bf16.h>

typedef __attribute__((ext_vector_type(2))) float v2f;
typedef __attribute__((ext_vector_type(8))) float v8f;

#define LAG_C 0.5f

#define WMMA_F32(a, b, c) \
  __builtin_amdgcn_wmma_f32_16x16x4_f32(false, (a), false, (b), (short)0, (c), false, false)

// ---------------------------------------------------------------------------
// One wave computes a 16(M) x 64(N) strip of C = A[M,K] * Bw[N,K]^T with
// V_WMMA_F32_16X16X4_F32. 2-deep software pipeline so ~10 global_load_b64
// stay in flight across each 4-WMMA burst (avoids s_wait_loadcnt 0x0 stalls).
//
// VGPR layouts (CDNA5 ISA 7.12.2):
//   A (16x4 f32):  (m,k): lane = (k>>1)*16 + m, vgpr = k&1  -> float2 per lane
//   B (4x16 f32):  mirrored with N across lanes 0..15
//   C/D (16x16):   m = v + 8*(lane>=16), n = lane&15
// Requires K % 8 == 0.
// ---------------------------------------------------------------------------
__device__ __forceinline__ void wmma_rowstrip4_f32(
    const float* __restrict__ A, int lda,
    const float* __restrict__ Bw, int ldb,
    int rowBase, int colBase, int K,
    unsigned lane, v8f c[4])
{
  const int half = (lane >> 4) & 1;
  const int l16  = (int)(lane & 15u);

  const float* ap = A + (size_t)(rowBase + l16) * lda + half * 2;
  const float* b0 = Bw + (size_t)(colBase +  0 + l16) * ldb + half * 2;
  const float* b1 = Bw + (size_t)(colBase + 16 + l16) * ldb + half * 2;
  const float* b2 = Bw + (size_t)(colBase + 32 + l16) * ldb + half * 2;
  const float* b3 = Bw + (size_t)(colBase + 48 + l16) * ldb + half * 2;

  v8f z = {0.f, 0.f, 0.f, 0.f, 0.f, 0.f, 0.f, 0.f};
  c[0] = z; c[1] = z; c[2] = z; c[3] = z;

  // Pipeline stages 0 (k) and 1 (k+4) pre-loaded.
  v2f a0 = *(const v2f*)(ap);     v2f a1 = *(const v2f*)(ap + 4);
  v2f p0 = *(const v2f*)(b0);     v2f p1 = *(const v2f*)(b0 + 4);
  v2f q0 = *(const v2f*)(b1);     v2f q1 = *(const v2f*)(b1 + 4);
  v2f r0 = *(const v2f*)(b2);     v2f r1 = *(const v2f*)(b2 + 4);
  v2f s0 = *(const v2f*)(b3);     v2f s1 = *(const v2f*)(b3 + 4);

  int k = 0;
  for (; k < K - 8; k += 4) {
    // Prefetch stage k+8 before consuming stage k.
    v2f a2 = *(const v2f*)(ap + k + 8);
    v2f p2 = *(const v2f*)(b0 + k + 8);
    v2f q2 = *(const v2f*)(b1 + k + 8);
    v2f r2 = *(const v2f*)(b2 + k + 8);
    v2f s2 = *(const v2f*)(b3 + k + 8);
    c[0] = WMMA_F32(a0, p0, c[0]);
    c[1] = WMMA_F32(a0, q0, c[1]);
    c[2] = WMMA_F32(a0, r0, c[2]);
    c[3] = WMMA_F32(a0, s0, c[3]);
    a0 = a1; p0 = p1; q0 = q1; r0 = r1; s0 = s1;
    a1 = a2; p1 = p2; q1 = q2; r1 = r2; s1 = s2;
  }
  // Drain the two in-flight stages.
  c[0] = WMMA_F32(a0, p0, c[0]);
  c[1] = WMMA_F32(a0, q0, c[1]);
  c[2] = WMMA_F32(a0, r0, c[2]);
  c[3] = WMMA_F32(a0, s0, c[3]);
  c[0] = WMMA_F32(a1, p1, c[0]);
  c[1] = WMMA_F32(a1, q1, c[1]);
  c[2] = WMMA_F32(a1, r1, c[2]);
  c[3] = WMMA_F32(a1, s1, c[3]);
}

// ---------------------------------------------------------------------------
// C[M,N] = A[M,K] @ Bw[N,K]^T + bias1[N] (+ bias2[N] if non-null)
// One wave per 16x64 output strip; blockDim = 256 (8 waves).
// ---------------------------------------------------------------------------
__global__ void gemm_nt_bias2_f32(const float* __restrict__ A,
                                  const float* __restrict__ Bw,
                                  const float* __restrict__ bias1,
                                  const float* __restrict__ bias2,
                                  float* __restrict__ C,
                                  int M, int N, int K)
{
  const unsigned lane = threadIdx.x & 31u;
  const unsigned wave = blockIdx.x * (blockDim.x >> 5) + (threadIdx.x >> 5);
  const unsigned nSuper = (unsigned)N >> 6;
  const int rowBase = (int)(wave / nSuper) * 16;
  const int colBase = (int)(wave % nSuper) * 64;
  if (rowBase >= M) return;  // wave-uniform (grid sized exactly; EXEC stays full)

  v8f c[4];
  wmma_rowstrip4_f32(A, K, Bw, K, rowBase, colBase, K, lane, c);

  const int half = (lane >> 4) & 1;
  const int l16  = (int)(lane & 15u);
#pragma unroll
  for (int j = 0; j < 4; ++j) {
    const int n = colBase + j * 16 + l16;
    const float bv = bias1[n] + (bias2 ? bias2[n] : 0.0f);
#pragma unroll
    for (int v = 0; v < 8; ++v) {
      const int m = rowBase + v + half * 8;
      C[(size_t)m * N + n] = c[j][v] + bv;
    }
  }
}

__global__ void zero_u32_kernel(unsigned* p) { *p = 0u; }

// ---------------------------------------------------------------------------
// Persistent scan kernel: all T timesteps in ONE launch.
//   pre   = X[:,t,:] + h @ W_h^T     (X pre-holds obs@W_in^T + b_in + b_h)
//   h_new = LAG*relu(pre) + (1-LAG)*h ;  hs[:,t,:] = h_new
// Grid is exactly 8 blocks x 8 waves = 64 waves = (B/16)*(H/64) strips.
// Timesteps are separated by a device-wide atomic counter barrier
// (monotonic target (t+1)*gridDim.x; counter zeroed per call by init kernel).
// ---------------------------------------------------------------------------
__global__ void rnn_scan_persistent(const float* __restrict__ X,    // [B,T,H]
                                    const float* __restrict__ W_h,  // [H,H]
                                    float* __restrict__ h_a,        // [B,H]
                                    float* __restrict__ h_b,        // [B,H]
                                    float* __restrict__ hs_out,     // [B,T,H]
                                    unsigned* __restrict__ bar_cnt,
                                    int B, int T, int H)
{
  const unsigned lane = threadIdx.x & 31u;
  const unsigned wave = blockIdx.x * (blockDim.x >> 5) + (threadIdx.x >> 5);
  const unsigned nSuper = (unsigned)H >> 6;
  const int rowBase = (int)(wave / nSuper) * 16;
  const int colBase = (int)(wave % nSuper) * 64;
  const int half = (lane >> 4) & 1;
  const int l16  = (int)(lane & 15u);
  const unsigned nBlocks = gridDim.x;

  float* bufs[2] = {h_a, h_b};

  for (int t = 0; t < T; ++t) {
    const float* hp = bufs[t & 1];
    float*       hn = bufs[(t + 1) & 1];

    v8f c[4];
    wmma_rowstrip4_f32(hp, H, W_h, H, rowBase, colBase, H, lane, c);

#pragma unroll
    for (int j = 0; j < 4; ++j) {
      const int n = colBase + j * 16 + l16;
#pragma unroll
      for (int v = 0; v < 8; ++v) {
        const int m  = rowBase + v + half * 8;
        const size_t bt = (size_t)m * T * H + (size_t)t * H + n;
        const float pre = c[j][v] + X[bt];
        const float hpv = hp[(size_t)m * H + n];
        const float hv  = LAG_C * fmaxf(pre, 0.0f) + (1.0f - LAG_C) * hpv;
        hn[(size_t)m * H + n] = hv;
        hs_out[bt] = hv;
      }
    }

    // ---- device-wide barrier between timesteps ----
    __syncthreads();
    if (threadIdx.x == 0) {
      __threadfence();  // release: flush h_new stores device-wide
      __hip_atomic_fetch_add(bar_cnt, 1u, __ATOMIC_ACQ_REL, __HIP_MEMORY_SCOPE_AGENT);
      const unsigned target = (unsigned)(t + 1) * nBlocks;
      while (__hip_atomic_load(bar_cnt, __ATOMIC_ACQUIRE, __HIP_MEMORY_SCOPE_AGENT) < target) {
        __builtin_amdgcn_s_sleep(8);
      }
    }
    __syncthreads();
    __threadfence();  // acquire: invalidate stale WGP-local lines for all waves
  }
}

extern "C" void kernel_launch(void* const* d_in, const int* in_sizes, int n_in,
                              void* d_out, int out_size, void* d_ws, size_t ws_size,
                              hipStream_t stream) {
  const float* obs   = (const float*)d_in[0];  // [B,T,D]
  const float* h0    = (const float*)d_in[1];  // [B,H]
  const float* W_in  = (const float*)d_in[2];  // [H,D]
  const float* b_in  = (const float*)d_in[3];  // [H]
  const float* W_h   = (const float*)d_in[4];  // [H,H]
  const float* b_h   = (const float*)d_in[5];  // [H]
  const float* W_out = (const float*)d_in[6];  // [O,H]
  const float* b_out = (const float*)d_in[7];  // [O]

  const int B = 64, T = 512, D = 256, H = 1024, O = 256;

  float* outs = (float*)d_out;                      // [B,T,O]
  float* hs   = (float*)d_out + (size_t)B * T * O;  // [B,T,H]

  // Workspace: X [B,T,H] | h ping [B,H] | h pong [B,H] | barrier counter
  float*    X   = (float*)d_ws;
  float*    hb0 = X + (size_t)B * T * H;
  float*    hb1 = hb0 + (size_t)B * H;
  unsigned* bar = (unsigned*)(hb1 + (size_t)B * H);

  // Reset barrier counter + seed recurrent state (both graph-capture safe).
  zero_u32_kernel<<<1, 1, 0, stream>>>(bar);
  hipMemcpyAsync(hb0, h0, sizeof(float) * (size_t)B * H,
                 hipMemcpyDeviceToDevice, stream);

  // Phase 1: X = obs2d @ W_in^T + b_in + b_h   (M=32768, N=1024, K=256)
  {
    const int M = B * T, N = H, K = D;
    const unsigned waves = (unsigned)(M / 16) * (unsigned)(N / 64);  // 32768
    gemm_nt_bias2_f32<<<waves / 8, 256, 0, stream>>>(obs, W_in, b_in, b_h, X, M, N, K);
  }

  // Phase 2: whole sequential scan in one persistent launch (8 blocks).
  {
    const unsigned waves = (unsigned)(B / 16) * (unsigned)(H / 64);  // 64
    rnn_scan_persistent<<<waves / 8, 256, 0, stream>>>(X, W_h, hb0, hb1, hs, bar, B, T, H);
  }

  // Phase 3: outs = hs @ W_out^T + b_out   (M=32768, N=256, K=1024)
  {
    const int M = B * T, N = O, K = H;
    const unsigned waves = (unsigned)(M / 16) * (unsigned)(N / 64);  // 8192
    gemm_nt_bias2_f32<<<waves / 8, 256, 0, stream>>>(hs, W_out, b_out, nullptr, outs, M, N, K);
  }
}